// Router_86311662780524
// MI455X (gfx1250) — compile-verified
//
#include <hip/hip_runtime.h>
#include <math.h>

// MoE top-1 router (eval). N=8192 tokens, E=8 experts, Ccap=1536.
// Output (float, flat): dispatch[N*E*Ccap] | combine[N*E*Ccap] | z,aux,div,std
// Perf model: 805MB of output -> store-BW bound (~35us @ 23.3TB/s).
// Strategy: NT b128 zero-fill + sparse scatter; WMMA f32 16x16x4 computes the
// 8x8 Gram matrix for div_loss; all reductions fixed-order (deterministic).

#define NE 8            // experts
#define TPB 256

typedef float v2f __attribute__((ext_vector_type(2)));
typedef float v8f __attribute__((ext_vector_type(8)));
typedef float f4  __attribute__((ext_vector_type(4)));

__device__ __forceinline__ float clamp_temp(float x) {
    return fminf(fmaxf(x, -10.0f), 10.0f) * (1.0f / 1.5f);
}

// ---------------- Kernel A: per-token routing + partial sums -------------
// partials per block (19 floats): [0]=sum lse^2, [1..8]=sum p_e, [9..16]=count f_e,
// [17]=sum raw, [18]=sum raw^2
__global__ void kA_route(const float* __restrict__ logits, int N,
                         int* __restrict__ w_idx, float* __restrict__ w_prob,
                         float* __restrict__ partials) {
    __shared__ float red[TPB];
    const int tid = threadIdx.x;
    const int n = blockIdx.x * TPB + tid;
    const bool valid = (n < N);

    float raw[NE];
    float raw_s = 0.f, raw_q = 0.f;
    float l[NE];
    int idx = 0;
    float prob = 0.f, lse = 0.f, p[NE];

    if (valid) {
        const f4* row = (const f4*)(logits + (size_t)n * NE);
        f4 r0 = row[0], r1 = row[1];
        raw[0]=r0.x; raw[1]=r0.y; raw[2]=r0.z; raw[3]=r0.w;
        raw[4]=r1.x; raw[5]=r1.y; raw[6]=r1.z; raw[7]=r1.w;
        #pragma unroll
        for (int e = 0; e < NE; ++e) {
            raw_s += raw[e];
            raw_q += raw[e] * raw[e];
            l[e] = clamp_temp(raw[e]);
        }
        float m = l[0]; idx = 0;
        #pragma unroll
        for (int e = 1; e < NE; ++e) if (l[e] > m) { m = l[e]; idx = e; }
        float se = 0.f;
        #pragma unroll
        for (int e = 0; e < NE; ++e) { p[e] = expf(l[e] - m); se += p[e]; }
        float inv = 1.0f / se;
        #pragma unroll
        for (int e = 0; e < NE; ++e) p[e] *= inv;
        lse = m + logf(se);
        prob = p[idx];
        w_idx[n]  = idx;
        w_prob[n] = prob;
    } else {
        #pragma unroll
        for (int e = 0; e < NE; ++e) p[e] = 0.f;
        idx = -1;
    }

    float vals[19];
    vals[0] = valid ? lse * lse : 0.f;
    #pragma unroll
    for (int e = 0; e < NE; ++e) vals[1 + e] = p[e];
    #pragma unroll
    for (int e = 0; e < NE; ++e) vals[9 + e] = (idx == e) ? 1.f : 0.f;
    vals[17] = raw_s; vals[18] = raw_q;

    // fixed-order tree reductions -> deterministic
    for (int k = 0; k < 19; ++k) {
        red[tid] = vals[k];
        __syncthreads();
        #pragma unroll
        for (int s = TPB / 2; s > 0; s >>= 1) {
            if (tid < s) red[tid] += red[tid + s];
            __syncthreads();
        }
        if (tid == 0) partials[(size_t)blockIdx.x * 19 + k] = red[0];
        __syncthreads();
    }
}

// ---------------- Kernel B: WMMA Gram matrix + scalar losses -------------
// One block, 256 threads = 8 waves. G = Lc^T * Lc via V_WMMA_F32_16X16X4_F32.
// A(16x4: lane=M, vgprs=K) and B(4x16: lane=N, vgprs=K) both hold
// Lc[k0+2*(lane>>4)+v][lane&15] -> one v2f feeds both operands.
__global__ void kB_losses(const float* __restrict__ logits, int N, int nblk,
                          const float* __restrict__ partials,
                          float* __restrict__ out_scalars) {
    __shared__ float gsh[8][32][8];
    __shared__ float G[16][16];
    __shared__ float sums[19];

    const int tid  = threadIdx.x;
    const int wave = tid >> 5;
    const int lane = tid & 31;
    const int col  = lane & 15;
    const int kh   = lane >> 4;

    v8f c = {};
    const int chunks = N >> 2;            // 4 tokens (K) per WMMA
    for (int i = wave; i < chunks; i += 8) {
        const int t0 = i * 4 + 2 * kh;    // this half-wave's K pair
        v2f a;
        if (col < NE) {
            a.x = clamp_temp(logits[(size_t)t0 * NE + col]);
            a.y = clamp_temp(logits[(size_t)(t0 + 1) * NE + col]);
        } else {
            a.x = 0.f; a.y = 0.f;         // pad experts 8..15 with zeros
        }
        // D = A*B + C ; emits v_wmma_f32_16x16x4_f32
        c = __builtin_amdgcn_wmma_f32_16x16x4_f32(
                false, a, false, a, (short)0, c, false, false);
    }

    #pragma unroll
    for (int r = 0; r < 8; ++r) gsh[wave][lane][r] = c[r];
    __syncthreads();

    if (tid < 32) {                       // cross-wave reduce + de-swizzle C layout
        #pragma unroll
        for (int r = 0; r < 8; ++r) {
            float s = 0.f;
            #pragma unroll
            for (int w = 0; w < 8; ++w) s += gsh[w][tid][r];
            const int mm = (tid < 16) ? r : (r + 8);
            const int nn = (tid < 16) ? tid : (tid - 16);
            G[mm][nn] = s;
        }
    }
    if (tid == 0) {                       // serial -> deterministic
        for (int k = 0; k < 19; ++k) {
            float s = 0.f;
            for (int b = 0; b < nblk; ++b) s += partials[(size_t)b * 19 + k];
            sums[k] = s;
        }
    }
    __syncthreads();

    if (tid == 0) {
        const float Nf = (float)N;
        const float z_loss = sums[0] / Nf;
        float aux = 0.f;
        for (int e = 0; e < NE; ++e)
            aux += (sums[9 + e] / Nf) * (sums[1 + e] / Nf);
        aux *= (float)NE;
        const float M = Nf * (float)NE;   // unbiased std over raw logits
        const float var = (sums[18] - sums[17] * sums[17] / M) / (M - 1.0f);
        const float stdv = sqrtf(fmaxf(var, 0.f));
        float nrm[NE];
        for (int e = 0; e < NE; ++e) nrm[e] = fmaxf(sqrtf(G[e][e]), 1e-12f);
        float dv = 0.f;
        for (int i = 0; i < NE; ++i)
            for (int j = 0; j < NE; ++j)
                if (i != j) {
                    float v = G[i][j] / (nrm[i] * nrm[j]);
                    dv += v * v;
                }
        dv /= (float)(NE * (NE - 1));
        out_scalars[0] = z_loss;
        out_scalars[1] = aux;
        out_scalars[2] = dv;
        out_scalars[3] = stdv;
    }
}

// ---------------- Kernel Z: NT b128 zero-fill of dispatch+combine --------
__global__ void kZ_zero(f4* __restrict__ out4, long long n4) {
    const long long base = (long long)blockIdx.x * (TPB * 8) + threadIdx.x;
    const f4 z = {0.f, 0.f, 0.f, 0.f};
    #pragma unroll
    for (int j = 0; j < 8; ++j) {
        const long long i = base + (long long)j * TPB;
        if (i < n4) __builtin_nontemporal_store(z, &out4[i]);
    }
}

// ---------------- Kernel C: capacity rank + sparse scatter ---------------
// rank(n) = #{m : idx[m]==idx[n] && (prob[m]>prob[n] || (prob[m]==prob[n] && m<n))}
// == position in stable argsort(-prob) restricted to that expert.
__global__ void kC_scatter(const int* __restrict__ w_idx,
                           const float* __restrict__ w_prob,
                           int N, int Ccap, float* __restrict__ out) {
    __shared__ float sp[TPB];
    __shared__ int   si[TPB];
    const int tid = threadIdx.x;
    const int n = blockIdx.x * TPB + tid;
    const bool valid = (n < N);
    const float myp = valid ? w_prob[n] : 0.f;
    const int   myi = valid ? w_idx[n]  : -1;

    int rank = 0;
    for (int t0 = 0; t0 < N; t0 += TPB) {
        const int m = t0 + tid;
        sp[tid] = (m < N) ? w_prob[m] : 0.f;
        si[tid] = (m < N) ? w_idx[m]  : -2;
        __syncthreads();
        const int lim = min(TPB, N - t0);
        for (int j = 0; j < lim; ++j) {
            const int m2 = t0 + j;
            if (si[j] == myi && (sp[j] > myp || (sp[j] == myp && m2 < n)))
                ++rank;
        }
        __syncthreads();
    }

    if (valid && rank < Ccap) {          // unique slot per token -> no conflicts
        const long long EC  = (long long)NE * Ccap;
        const long long off = (long long)n * EC + (long long)myi * Ccap + rank;
        out[off] = 1.0f;                                  // dispatch
        out[(long long)N * EC + off] = myp;               // combine = gate
    }
}

// ---------------- Launch ----------------
extern "C" void kernel_launch(void* const* d_in, const int* in_sizes, int n_in,
                              void* d_out, int out_size, void* d_ws, size_t ws_size,
                              hipStream_t stream) {
    // inputs: d_in[0] = X (unused), d_in[1] = logits [N, 8]
    const float* logits = (const float*)d_in[1];
    const int N = in_sizes[1] / NE;                       // 8192
    int Ccap = (3 * N + 2 * NE - 1) / (2 * NE);           // ceil(1.5*N/E) = 1536
    if (Ccap < 1) Ccap = 1;

    float* out = (float*)d_out;
    const long long dense = 2LL * N * NE * Ccap;          // dispatch + combine
    float* out_scalars = out + dense;

    // workspace layout
    int*   w_idx    = (int*)d_ws;
    float* w_prob   = (float*)d_ws + N;
    float* partials = (float*)d_ws + 2 * (size_t)N;

    const int nblk = (N + TPB - 1) / TPB;                 // 32

    kA_route<<<nblk, TPB, 0, stream>>>(logits, N, w_idx, w_prob, partials);
    kB_losses<<<1, TPB, 0, stream>>>(logits, N, nblk, partials, out_scalars);

    const long long n4 = dense / 4;                       // dense % 4 == 0 (E=8)
    const long long zblk = (n4 + (long long)TPB * 8 - 1) / ((long long)TPB * 8);
    kZ_zero<<<(unsigned)zblk, TPB, 0, stream>>>((f4*)out, n4);

    kC_scatter<<<nblk, TPB, 0, stream>>>(w_idx, w_prob, N, Ccap, out);
}